// MyGCNConv_70566312673785
// MI455X (gfx1250) — compile-verified
//
#include <hip/hip_runtime.h>

#define F 256  // F_IN == F_OUT == 256

typedef __attribute__((ext_vector_type(2))) float v2f;
typedef __attribute__((ext_vector_type(8))) float v8f;

// --- deg[row[e]] += C[e] ------------------------------------------------
__global__ void deg_kernel(const int* __restrict__ row, const float* __restrict__ Cv,
                           float* __restrict__ deg, int E) {
    int e = blockIdx.x * blockDim.x + threadIdx.x;
    if (e < E) unsafeAtomicAdd(&deg[row[e]], Cv[e]);
}

// --- dinv = deg > 0 ? rsqrt(deg) : 0 -----------------------------------
__global__ void dinv_kernel(const float* __restrict__ deg, float* __restrict__ dinv, int N) {
    int i = blockIdx.x * blockDim.x + threadIdx.x;
    if (i < N) {
        float d = deg[i];
        dinv[i] = d > 0.f ? rsqrtf(d) : 0.f;
    }
}

// --- SpMM: prop[row[e]] += w_norm[e] * x[col[e]], one wave per edge ----
__global__ void __launch_bounds__(256)
spmm_kernel(const int* __restrict__ row, const int* __restrict__ col,
            const float* __restrict__ Cv, const float* __restrict__ dinv,
            const float* __restrict__ x, float* __restrict__ prop, int E) {
    int e = blockIdx.x * 8 + (threadIdx.x >> 5);   // 8 waves per block, 1 edge per wave
    if (e >= E) return;
    int lane = threadIdx.x & 31;
    int r = row[e], c = col[e];
    float w = dinv[r] * Cv[e] * dinv[c];
    const float4* xr = (const float4*)(x + (size_t)c * F);
    float* pr = prop + (size_t)r * F;
#pragma unroll
    for (int i = 0; i < 2; ++i) {
        int q = lane + i * 32;          // float4 index within the 256-f32 row
        float4 v = xr[q];
        float* p = pr + q * 4;
        unsafeAtomicAdd(p + 0, w * v.x);
        unsafeAtomicAdd(p + 1, w * v.y);
        unsafeAtomicAdd(p + 2, w * v.z);
        unsafeAtomicAdd(p + 3, w * v.w);
    }
}

// --- out = prop @ W^T + b via V_WMMA_F32_16X16X4_F32 -------------------
// One wave computes one 16x16 output tile; K loop of 64 fp32 WMMAs.
// A (16x4 f32): lanes 0-15 hold rows M=0..15 with K=0,1; lanes 16-31 K=2,3.
// B (4x16 f32): lanes hold N across lanes; VGPR0/1 = K pair. B[k][n]=W[n][k],
// so each lane reads a contiguous float2 from row (n0+r) of W.
// C/D (16x16 f32, 8 VGPRs): lanes 0-15 -> M=v, lanes 16-31 -> M=8+v, N=lane&15.
__global__ void __launch_bounds__(256)
gemm_wmma_kernel(const float* __restrict__ prop, const float* __restrict__ W,
                 const float* __restrict__ bias, float* __restrict__ out,
                 int mTiles) {
    int mt = blockIdx.y * blockDim.y + threadIdx.y;  // this wave's M tile
    if (mt >= mTiles) return;
    int n0   = blockIdx.x * 16;
    int m0   = mt * 16;
    int lane = threadIdx.x;        // 0..31 (wave32)
    int h    = lane >> 4;          // half-wave
    int r    = lane & 15;

    const float* aRow = prop + (size_t)(m0 + r) * F + 2 * h;
    const float* bRow = W    + (size_t)(n0 + r) * F + 2 * h;

    v8f c = {0.f, 0.f, 0.f, 0.f, 0.f, 0.f, 0.f, 0.f};
#pragma unroll 4
    for (int k = 0; k < F; k += 4) {
        v2f a = *(const v2f*)(aRow + k);
        v2f b = *(const v2f*)(bRow + k);
        // 8 args: (neg_a, A, neg_b, B, c_mod, C, reuse_a, reuse_b)
        c = __builtin_amdgcn_wmma_f32_16x16x4_f32(false, a, false, b,
                                                  (short)0, c, false, false);
    }

    float bv = bias[n0 + r];
    float* o = out + ((size_t)m0 + (size_t)h * 8) * F + n0 + r;
#pragma unroll
    for (int v = 0; v < 8; ++v)
        o[(size_t)v * F] = c[v] + bv;
}

extern "C" void kernel_launch(void* const* d_in, const int* in_sizes, int n_in,
                              void* d_out, int out_size, void* d_ws, size_t ws_size,
                              hipStream_t stream) {
    const float* x   = (const float*)d_in[0];   // [N, 256]
    const int*   ei  = (const int*)d_in[1];     // [2, E]
    const float* Cv  = (const float*)d_in[2];   // [E]
    const float* W   = (const float*)d_in[3];   // [256, 256]
    const float* b   = (const float*)d_in[4];   // [256]
    float*       out = (float*)d_out;           // [N, 256]

    int N = in_sizes[0] / F;
    int E = in_sizes[2];
    const int* row = ei;
    const int* col = ei + E;

    // workspace layout: deg[N] | dinv[N] | prop[N*F]
    float* deg  = (float*)d_ws;
    float* dinv = deg + N;
    float* prop = dinv + N;

    size_t zero_bytes = ((size_t)2 * N + (size_t)N * F) * sizeof(float);
    hipMemsetAsync(d_ws, 0, zero_bytes, stream);

    deg_kernel <<<(E + 255) / 256, 256, 0, stream>>>(row, Cv, deg, E);
    dinv_kernel<<<(N + 255) / 256, 256, 0, stream>>>(deg, dinv, N);
    spmm_kernel<<<(E + 7) / 8, 256, 0, stream>>>(row, col, Cv, dinv, x, prop, E);

    int mTiles = N / 16;                      // N = 50000 -> 3125 exact tiles
    dim3 grid(F / 16, (mTiles + 7) / 8);      // (16, 391)
    dim3 block(32, 8);                        // 8 waves/block, 1 tile/wave
    gemm_wmma_kernel<<<grid, block, 0, stream>>>(prop, W, b, out, mTiles);
}